// SNNReadoutLayer_24292335026510
// MI455X (gfx1250) — compile-verified
//
#include <hip/hip_runtime.h>
#include <stdint.h>

// SNN readout layer, fused GEMM + temporal scan for MI455X (gfx1250).
//
//   h[b,t,o] = sum_i x[b,t,i] * W[o,i]        (fp32 WMMA 16x16x4)
//   out[t]   = BETA*out[t-1] + flt[t-1]
//   flt[t]   = ALPHA*flt[t-1] + h[t]
//
// One block = (b, 64-wide o slice); 4 waves, each wave owns 16 o-channels and
// keeps the scan state in registers across the whole T loop. h never touches
// HBM. The x-tile (A matrix) is streamed into a double-buffered LDS tile by
// the Tensor Data Mover one chunk ahead of the WMMA consumers.

typedef __attribute__((ext_vector_type(2))) float v2f;
typedef __attribute__((ext_vector_type(8))) float v8f;
typedef __attribute__((ext_vector_type(4))) unsigned int v4u;
typedef __attribute__((ext_vector_type(8))) int v8i;
typedef __attribute__((ext_vector_type(4))) int v4i;

constexpr int   kB = 64;
constexpr int   kT = 1000;
constexpr int   kI = 512;
constexpr int   kO = 256;
constexpr int   kChunks = (kT + 15) / 16;   // 63
constexpr float kAlpha = 0.95f;
constexpr float kBeta  = 0.90f;

// TDM pad engine: pad_interval=7 (every 256 dwords) + pad_amount=1 (2 dwords)
// => row stride 516 dwords (516 % 64 == 4 -> conflict-free column reads),
// element (r, i) lives at dword r*516 + i + 2*(i >= 256).
constexpr int kRowStride = kI + 4;          // 516 dwords per 512-float row

__global__ __launch_bounds__(128) void snn_readout_fused(
    const float* __restrict__ x,   // [B][T][I]
    const float* __restrict__ W,   // [O][I]
    float* __restrict__ out)       // [B][T][O]
{
  __shared__ float lds_a[2][16 * kRowStride];  // 66 KB double-buffered A tile
  __shared__ float scanbuf[4][16][16];         //  4 KB per-wave C staging

  const int tid  = threadIdx.x;
  const int wave = tid >> 5;
  const int lane = tid & 31;
  const int half = lane >> 4;            // 0: lanes 0-15, 1: lanes 16-31
  const int l16  = lane & 15;

  const int b      = blockIdx.x >> 2;          // 64 b values
  const int o_base = (blockIdx.x & 3) * 64;    // 4 o-slices of 64
  const int o_wave = o_base + wave * 16;       // this wave's 16 channels

  const float* __restrict__ xb = x + (size_t)b * kT * kI;
  // WMMA f32 16x16x4 operand layout (ISA 7.12.2): A: M = lane%16,
  // K = vgpr + 2*(lane>=16); B mirrored with N = lane%16. Consecutive K per
  // lane -> natural 8-byte (b64) loads for both fragments.
  const float* __restrict__ wrow = W + (size_t)(o_wave + l16) * kI + 2 * half;

  // Flat->LDS: addr[31:0] of a flat shared pointer is the wave-relative LDS
  // byte address (ISA 10.2), which is what D#.lds_addr wants.
  const unsigned ldsAddr[2] = {
      (unsigned)(uintptr_t)&lds_a[0][0],
      (unsigned)(uintptr_t)&lds_a[1][0]};
  const uint64_t gbase = (uint64_t)(uintptr_t)xb;

  // Issue one TDM transfer: x[b, t0:t0+16, 0:512] -> lds_a[buf] (wave op).
  auto tdm_issue = [&](int buf, int t0) {
    const unsigned rows = (kT - t0 < 16) ? (unsigned)(kT - t0) : 16u;  // OOB rows -> 0
    const uint64_t ga   = gbase + (uint64_t)t0 * kI * 4u;
    v4u g0;
    g0[0] = 1u;                                   // count=1, user descriptor
    g0[1] = ldsAddr[buf];                         // lds_addr (bytes)
    g0[2] = (unsigned)ga;                         // global_addr[31:0]
    g0[3] = (unsigned)(ga >> 32) | (2u << 30);    // global_addr[56:32] | type=2
    v8i g1;
    g1[0] = (2 << 16)        // data_size = 4 bytes
          | (1 << 20)        // pad_enable
          | (7 << 22)        // pad_interval: 256 dwords
          | (1 << 25);       // pad_amount: 2 dwords
    g1[1] = (kI & 0xFFFF) << 16;          // tensor_dim0[15:0] = 512
    g1[2] = (int)(rows << 16);            // tensor_dim0[31:16]=0 | tensor_dim1[15:0]
    g1[3] = kI << 16;                     // tensor_dim1[31:16]=0 | tile_dim0 = 512
    g1[4] = 16;                           // tile_dim1 = 16, tile_dim2 = 0
    g1[5] = kI;                           // tensor_dim0_stride[31:0] = 512
    g1[6] = 0;                            // stride hi / tensor_dim1_stride lo
    g1[7] = 0;                            // tensor_dim1_stride hi
    const v4i z4 = {0, 0, 0, 0};          // groups 2/3 unused (2D tile)
    const v8i z8 = {0, 0, 0, 0, 0, 0, 0, 0};
    __builtin_amdgcn_tensor_load_to_lds(g0, g1, z4, z4, z8, 0);
  };

  // Scan state: lanes 0..15 of each wave own channel o = o_wave + lane.
  float flt = 0.0f;
  float acc = 0.0f;

  if (wave == 0) tdm_issue(0, 0);  // prefetch chunk 0

  for (int ci = 0; ci < kChunks; ++ci) {
    const int t0  = ci * 16;
    const int cur = ci & 1;

    if (wave == 0) {
      if (ci + 1 < kChunks) {
        tdm_issue(cur ^ 1, t0 + 16);             // prefetch next chunk
        __builtin_amdgcn_s_wait_tensorcnt(1);    // current chunk's DMA done
      } else {
        __builtin_amdgcn_s_wait_tensorcnt(0);
      }
    }
    __syncthreads();  // publish lds_a[cur]; also fences last chunk's reads

    // ---- GEMM: 16t x 16o tile, K = 512 in steps of 4.
    const float* __restrict__ abase = &lds_a[cur][0] + l16 * kRowStride + 2 * half;
    v8f c = {};
#pragma unroll 8
    for (int k = 0; k < 64; ++k) {               // columns 0..255
      v2f a  = *reinterpret_cast<const v2f*>(abase + 4 * k);      // ds b64
      v2f bf = *reinterpret_cast<const v2f*>(wrow + 4 * k);       // global b64
      c = __builtin_amdgcn_wmma_f32_16x16x4_f32(
          false, a, false, bf, (short)0, c, false, false);
    }
#pragma unroll 8
    for (int k = 64; k < 128; ++k) {             // columns 256..511 (+2 pad)
      v2f a  = *reinterpret_cast<const v2f*>(abase + 4 * k + 2);
      v2f bf = *reinterpret_cast<const v2f*>(wrow + 4 * k);
      c = __builtin_amdgcn_wmma_f32_16x16x4_f32(
          false, a, false, bf, (short)0, c, false, false);
    }

    // ---- Spill C to per-wave LDS tile: vgpr r, lane l -> t = r + 8*(l>=16),
    // o = l%16.
#pragma unroll
    for (int r = 0; r < 8; ++r) {
      scanbuf[wave][r + 8 * half][l16] = c[r];
    }
    __syncthreads();  // C visible; all lds_a[cur] reads drained before next TDM

    // ---- Sequential scan over the 16 timesteps of this chunk.
    if (lane < 16) {
      float* __restrict__ orow =
          out + ((size_t)b * kT + t0) * kO + o_wave + lane;
      const int tmax = (kT - t0 < 16) ? (kT - t0) : 16;
      for (int tt = 0; tt < tmax; ++tt) {
        const float h = scanbuf[wave][tt][lane];
        acc = kBeta * acc + flt;     // uses OLD flt (matches reference)
        flt = kAlpha * flt + h;
        orow[(size_t)tt * kO] = acc;
      }
    }
  }
}

extern "C" void kernel_launch(void* const* d_in, const int* in_sizes, int n_in,
                              void* d_out, int out_size, void* d_ws, size_t ws_size,
                              hipStream_t stream) {
  (void)in_sizes; (void)n_in; (void)d_ws; (void)ws_size; (void)out_size;
  const float* x = (const float*)d_in[0];   // inputs [64][1000][512] f32
  const float* W = (const float*)d_in[1];   // W      [256][512]      f32
  float* out     = (float*)d_out;           // out    [64][1000][256] f32

  dim3 grid(kB * (kO / 64));   // 256 blocks: (b, o-slice)
  dim3 block(128);             // 4 waves
  snn_readout_fused<<<grid, block, 0, stream>>>(x, W, out);
}